// GraphAttentionLayer_27960237097255
// MI455X (gfx1250) — compile-verified
//
#include <hip/hip_runtime.h>
#include <hip/hip_bf16.h>

// Shapes (compile-time constants from the reference)
#define BB 2
#define NN 384
#define EN 128
#define EE 64
#define HH 8
#define FF 256
#define INV_SQRT32 0.17677669529663687f

#define TW 8                       // (i,j) tile width for the edge kernel
#define NT (NN / TW)               // 48 tiles per dim
#define NPAIR (NT * (NT + 1) / 2)  // 1176 unordered tile pairs per batch

typedef float v2f __attribute__((ext_vector_type(2)));
typedef float v8f __attribute__((ext_vector_type(8)));

__device__ __forceinline__ float elu1(float x) {
    return x > 0.f ? x : __expf(x) - 1.0f;
}

// ---------------------------------------------------------------------------
// K1: nf = nodes @ W_node + b_node   ([768,128] x [128,256]) via f32 WMMA
// grid (48,16), 32 threads (1 wave) per block; 16x16 out tile, K=128 (32 steps)
// ---------------------------------------------------------------------------
__global__ __launch_bounds__(32) void k_nf(const float* __restrict__ nodes,
                                           const float* __restrict__ Wn,
                                           const float* __restrict__ bn,
                                           float* __restrict__ nf) {
    const int lane = threadIdx.x;
    const int m    = lane & 15;
    const int hi   = lane >> 4;       // 0: K 0,1   1: K 2,3
    const int koff = hi * 2;
    const int row  = blockIdx.x * 16 + m;   // A row for this lane
    const int col  = blockIdx.y * 16 + m;   // B/D col for this lane
    const float* arow = nodes + row * EN;

    v8f acc = {};
#pragma unroll
    for (int k = 0; k < EN / 4; ++k) {
        v2f a = *(const v2f*)(arow + 4 * k + koff);
        v2f b;
        b.x = Wn[(4 * k + koff) * FF + col];
        b.y = Wn[(4 * k + koff + 1) * FF + col];
        acc = __builtin_amdgcn_wmma_f32_16x16x4_f32(false, a, false, b,
                                                    (short)0, acc, false, false);
    }
#pragma unroll
    for (int r = 0; r < 8; ++r) {
        int orow = blockIdx.x * 16 + r + hi * 8;   // D row = r (+8 for hi lanes)
        nf[orow * FF + col] = acc[r] + bn[col];
    }
}

// ---------------------------------------------------------------------------
// K2: q = nf @ W_att    [768,8], K=256 (tiny)
// ---------------------------------------------------------------------------
__global__ __launch_bounds__(32) void k_q(const float* __restrict__ nf,
                                          const float* __restrict__ Watt,
                                          float* __restrict__ q) {
    int row = blockIdx.x;
    int t = threadIdx.x;
    if (t < HH) {
        const float* r = nf + row * FF;
        float acc = 0.f;
        for (int k = 0; k < FF; ++k) acc += r[k] * Watt[k * HH + t];
        q[row * HH + t] = acc;
    }
}

// ---------------------------------------------------------------------------
// K3: M = W_edge @ W_att [64,8];  c8 = b_edge @ W_att [8]
// ---------------------------------------------------------------------------
__global__ __launch_bounds__(512) void k_m(const float* __restrict__ We,
                                           const float* __restrict__ be,
                                           const float* __restrict__ Watt,
                                           float* __restrict__ Mm,
                                           float* __restrict__ c8) {
    int t = threadIdx.x;               // 512 = 64*8
    int c = t >> 3, h = t & 7;
    float acc = 0.f;
    for (int f = 0; f < FF; ++f) acc += We[c * FF + f] * Watt[f * HH + h];
    Mm[t] = acc;
    if (t < HH) {
        float a2 = 0.f;
        for (int f = 0; f < FF; ++f) a2 += be[f] * Watt[f * HH + t];
        c8[t] = a2;
    }
}

// ---------------------------------------------------------------------------
// K4: aw_n = softmax_j( (q_i . q_j)/sqrt(32) * mask_i )
// grid = B*N (row i), block = 384 (one j per thread)
// ---------------------------------------------------------------------------
__global__ __launch_bounds__(384) void k_awn(const float* __restrict__ q,
                                             const unsigned char* __restrict__ mask,
                                             float* __restrict__ awn) {
    __shared__ float qi[HH];
    __shared__ float red[512];
    int bi = blockIdx.x;               // b*N + i
    int b = bi / NN;
    int j = threadIdx.x;
    if (j < HH) qi[j] = q[bi * HH + j];
    __syncthreads();

    const float* qj = q + (b * NN + j) * HH;
    float s = 0.f;
#pragma unroll
    for (int h = 0; h < HH; ++h) s += qi[h] * qj[h];
    float mval = mask[bi] ? 1.0f : 0.0f;
    s *= INV_SQRT32 * mval;

    red[j] = s; __syncthreads();
    for (int st = 256; st > 0; st >>= 1) {
        if (j < st && j + st < NN) red[j] = fmaxf(red[j], red[j + st]);
        __syncthreads();
    }
    float mx = red[0]; __syncthreads();
    float e = __expf(s - mx);
    red[j] = e; __syncthreads();
    for (int st = 256; st > 0; st >>= 1) {
        if (j < st && j + st < NN) red[j] += red[j + st];
        __syncthreads();
    }
    awn[(size_t)bi * NN + j] = e / red[0];
}

// ---------------------------------------------------------------------------
// K5: aw_e = softmax_j( (edges[b,i,j,:] . (M q_i) + q_i.c8)/sqrt(32) * mask_i )
//     g[b,i,c] = sum_j aw_e[b,i,j] * edges[b,i,j,c]     (for agg_edge)
// grid = B*N, block = 384
// ---------------------------------------------------------------------------
__global__ __launch_bounds__(384) void k_awe(const float* __restrict__ q,
                                             const float* __restrict__ Mm,
                                             const float* __restrict__ c8,
                                             const float* __restrict__ edges,
                                             const unsigned char* __restrict__ mask,
                                             float* __restrict__ awe,
                                             float* __restrict__ g) {
    __shared__ float v[EE];
    __shared__ float qcs;
    __shared__ float aw[NN];
    __shared__ float red[512];
    int bi = blockIdx.x;
    int j = threadIdx.x;

    if (j < EE) {
        float acc = 0.f;
#pragma unroll
        for (int h = 0; h < HH; ++h) acc += Mm[j * HH + h] * q[bi * HH + h];
        v[j] = acc;
    }
    if (j == 0) {
        float acc = 0.f;
#pragma unroll
        for (int h = 0; h < HH; ++h) acc += c8[h] * q[bi * HH + h];
        qcs = acc;
    }
    __syncthreads();

    const float* erow = edges + ((size_t)bi * NN + j) * EE;
    float s = qcs;
#pragma unroll 16
    for (int c = 0; c < EE; ++c) s += erow[c] * v[c];
    float mval = mask[bi] ? 1.0f : 0.0f;
    s *= INV_SQRT32 * mval;

    red[j] = s; __syncthreads();
    for (int st = 256; st > 0; st >>= 1) {
        if (j < st && j + st < NN) red[j] = fmaxf(red[j], red[j + st]);
        __syncthreads();
    }
    float mx = red[0]; __syncthreads();
    float e = __expf(s - mx);
    red[j] = e; __syncthreads();
    for (int st = 256; st > 0; st >>= 1) {
        if (j < st && j + st < NN) red[j] += red[j + st];
        __syncthreads();
    }
    float w = e / red[0];
    awe[(size_t)bi * NN + j] = w;
    aw[j] = w;
    __syncthreads();

    if (j < EE) {   // column-owner reduction; edges row is L2-resident
        const float* ebase = edges + (size_t)bi * NN * EE + j;
        float acc = 0.f;
        for (int jj = 0; jj < NN; ++jj) acc += aw[jj] * ebase[(size_t)jj * EE];
        g[bi * EE + j] = acc;
    }
}

// ---------------------------------------------------------------------------
// K6: edge output. One block per unordered 8x8 (i,j) tile pair.
// Each wave owns one WMMA M-tile of 16 rows = 8 mirrored (i,j)/(j,i) pairs,
// recomputes ef = edges@W_edge + b_edge with f32 WMMA (K=64), applies
// ue = ef*(1+aw_e) + aw_n*nf, elu, averages mirrored rows in-register,
// and writes both (i,j) and (j,i) outputs.
// ---------------------------------------------------------------------------
__global__ __launch_bounds__(256) void k_edges(const float* __restrict__ edges,
                                               const float* __restrict__ We,
                                               const float* __restrict__ be,
                                               const float* __restrict__ nf,
                                               const float* __restrict__ awn,
                                               const float* __restrict__ awe,
                                               float* __restrict__ outE) {
    extern __shared__ float lds[];
    float* Wl    = lds;               // [64][256] k-major : 16384
    float* bel   = Wl + EE * FF;      // 256
    float* nfi   = bel + FF;          // [8][256]
    float* nfj   = nfi + TW * FF;     // [8][256]
    float* ae_ij = nfj + TW * FF;     // 64 each
    float* ae_ji = ae_ij + TW * TW;
    float* an_ij = ae_ji + TW * TW;
    float* an_ji = an_ij + TW * TW;

    int bp = blockIdx.x;
    int b  = bp / NPAIR;
    int p  = bp % NPAIR;
    int ti = 0;
    while (p >= NT - ti) { p -= NT - ti; ++ti; }
    int tj = ti + p;

    int tid = threadIdx.x;
    for (int idx = tid; idx < EE * FF; idx += 256) Wl[idx] = We[idx];
    if (tid < FF) bel[tid] = be[tid];
    for (int idx = tid; idx < TW * FF; idx += 256) {
        int r = idx >> 8, c = idx & 255;
        nfi[idx] = nf[(b * NN + ti * TW + r) * FF + c];
        nfj[idx] = nf[(b * NN + tj * TW + r) * FF + c];
    }
    if (tid < TW * TW) {
        int li = tid >> 3, lj = tid & 7;
        int ii = ti * TW + li, jj = tj * TW + lj;
        ae_ij[tid] = awe[((size_t)b * NN + ii) * NN + jj];
        ae_ji[tid] = awe[((size_t)b * NN + jj) * NN + ii];
        an_ij[tid] = awn[((size_t)b * NN + ii) * NN + jj];
        an_ji[tid] = awn[((size_t)b * NN + jj) * NN + ii];
    }
    __syncthreads();

    const int w    = tid >> 5;         // wave id = local i index
    const int lane = tid & 31;
    const int m    = lane & 15;
    const int hi   = lane >> 4;
    const int koff = hi * 2;

    // A rows: row 2q = (i, j_q), row 2q+1 = (j_q, i), i = ti*8+w, j_q = tj*8+q
    const int ii = ti * TW + w;
    {
        const int qrow = m >> 1, side = m & 1;
        const int jq = tj * TW + qrow;
        const int ri = side ? jq : ii;
        const int rj = side ? ii : jq;
        const float* arow = edges + (((size_t)b * NN + ri) * NN + rj) * EE;

        v2f a[16];
#pragma unroll
        for (int k = 0; k < 16; ++k) a[k] = *(const v2f*)(arow + 4 * k + koff);

        for (int nt = 0; nt < FF / 16; ++nt) {
            const int col = nt * 16 + m;
            v8f acc = {};
#pragma unroll
            for (int k = 0; k < 16; ++k) {
                v2f bv;
                bv.x = Wl[(4 * k + koff) * FF + col];
                bv.y = Wl[(4 * k + koff + 1) * FF + col];
                acc = __builtin_amdgcn_wmma_f32_16x16x4_f32(false, a[k], false, bv,
                                                            (short)0, acc, false, false);
            }
            // Epilogue: D row = r + hi*8 ; mirrored rows are (even,odd) pairs
#pragma unroll
            for (int r = 0; r < 8; r += 2) {
                int row0 = r + hi * 8;
                int q0   = row0 >> 1;           // combo within wave, 0..7
                int cidx = w * 8 + q0;
                int jg   = tj * TW + q0;
                float ef0 = acc[r]     + bel[col];
                float ef1 = acc[r + 1] + bel[col];
                float ue0 = ef0 * (1.0f + ae_ij[cidx]) + an_ij[cidx] * nfi[w * FF + col];
                float ue1 = ef1 * (1.0f + ae_ji[cidx]) + an_ji[cidx] * nfj[q0 * FF + col];
                float avg = 0.5f * (elu1(ue0) + elu1(ue1));
                outE[(((size_t)b * NN + ii) * NN + jg) * FF + col] = avg;
                outE[(((size_t)b * NN + jg) * NN + ii) * FF + col] = avg;
            }
        }
    }
}

// ---------------------------------------------------------------------------
// K7: node output.
// un[b,n,k] = nf + sum_i awn[b,i,n]*nf[b,i,k] + (g[b,n,:]@W_edge + b_edge)[k]
// ---------------------------------------------------------------------------
__global__ __launch_bounds__(256) void k_nodes(const float* __restrict__ nf,
                                               const float* __restrict__ awn,
                                               const float* __restrict__ g,
                                               const float* __restrict__ We,
                                               const float* __restrict__ be,
                                               float* __restrict__ outN) {
    int bn = blockIdx.x;              // b*N + n
    int b  = bn / NN;
    int n  = bn % NN;
    int k  = threadIdx.x;

    float acc = nf[bn * FF + k] + be[k];
    const float* nfb   = nf + b * NN * FF;
    const float* awcol = awn + (size_t)b * NN * NN + n;   // awn[b,i,n]
    for (int i = 0; i < NN; ++i)
        acc += awcol[(size_t)i * NN] * nfb[i * FF + k];
    const float* gr = g + bn * EE;
    for (int c = 0; c < EE; ++c)
        acc += gr[c] * We[c * FF + k];
    outN[bn * FF + k] = elu1(acc);
}

// ---------------------------------------------------------------------------
extern "C" void kernel_launch(void* const* d_in, const int* in_sizes, int n_in,
                              void* d_out, int out_size, void* d_ws, size_t ws_size,
                              hipStream_t stream) {
    const float*         nodes = (const float*)d_in[0];
    const float*         edges = (const float*)d_in[1];
    const unsigned char* mask  = (const unsigned char*)d_in[2];   // numpy bool
    const float*         Wn    = (const float*)d_in[3];
    const float*         bn    = (const float*)d_in[4];
    const float*         We    = (const float*)d_in[5];
    const float*         bev   = (const float*)d_in[6];
    const float*         Watt  = (const float*)d_in[7];

    float* ws  = (float*)d_ws;
    float* nf  = ws;                   // [2,384,256]   196608
    float* q   = nf  + BB * NN * FF;   // [2,384,8]     6144
    float* Mm  = q   + BB * NN * HH;   // [64,8]        512
    float* c8  = Mm  + EE * HH;        // [8] (pad 64)
    float* awn = c8  + 64;             // [2,384,384]   294912
    float* awe = awn + BB * NN * NN;   // [2,384,384]   294912
    float* g   = awe + BB * NN * NN;   // [2,384,64]    49152

    float* outN = (float*)d_out;
    float* outE = outN + BB * NN * FF;

    k_nf   <<<dim3((BB * NN) / 16, FF / 16), 32, 0, stream>>>(nodes, Wn, bn, nf);
    k_q    <<<BB * NN, 32, 0, stream>>>(nf, Watt, q);
    k_m    <<<1, 512, 0, stream>>>(We, bev, Watt, Mm, c8);
    k_awn  <<<BB * NN, NN, 0, stream>>>(q, mask, awn);
    k_awe  <<<BB * NN, NN, 0, stream>>>(q, Mm, c8, edges, mask, awe, g);

    size_t ldsBytes = (EE * FF + FF + 2 * TW * FF + 4 * TW * TW) * sizeof(float); // ~84 KB
    k_edges<<<BB * NPAIR, 256, ldsBytes, stream>>>(edges, We, bev, nf, awn, awe, outE);
    k_nodes<<<BB * NN, FF, 0, stream>>>(nf, awn, g, We, bev, outN);
}